// GCNBackbone_52312701665402
// MI455X (gfx1250) — compile-verified
//
#include <hip/hip_runtime.h>
#include <hip/hip_bf16.h>

typedef __attribute__((ext_vector_type(2))) float v2f;
typedef __attribute__((ext_vector_type(8))) float v8f;

#define NN 100000     // nodes (== 6250 * 16, exact)
#define EE 800000     // edges
#define DD 96         // feature dim (== 6 * 16, exact)
#define TILES_N (DD / 16)          // 6
#define TILES_M (NN / 16)          // 6250
#define TILES_TOTAL (TILES_M * TILES_N)  // 37500, divisible by 4

// ---------------- degree / normalization ----------------

__global__ __launch_bounds__(256) void k_init_deg(float* deg, int n) {
    int i = blockIdx.x * blockDim.x + threadIdx.x;
    if (i < n) deg[i] = 1.0f;  // self-loop contributes 1
}

__global__ __launch_bounds__(256) void k_deg_accum(const int* __restrict__ dst,
                                                   float* deg, int e) {
    int i = blockIdx.x * blockDim.x + threadIdx.x;
    if (i < e) atomicAdd(&deg[dst[i]], 1.0f);
}

__global__ __launch_bounds__(256) void k_rsqrt_inplace(float* deg, int n) {
    int i = blockIdx.x * blockDim.x + threadIdx.x;
    if (i < n) deg[i] = rsqrtf(deg[i]);
}

// ---------------- dense transform: H = X @ W via V_WMMA_F32_16X16X4_F32 ----------------
// One wave32 per 16x16 output tile. K=96 -> 24 chained WMMAs.
// A (16x4 f32) per-lane layout: lanes 0-15 hold M=lane, VGPR0=K0, VGPR1=K1;
// lanes 16-31 hold M=lane-16, VGPR0=K2, VGPR1=K3. B (4x16) symmetric across lanes.
__global__ __launch_bounds__(128) void k_gemm96_wmma(const float* __restrict__ X,
                                                     const float* __restrict__ W,
                                                     float* __restrict__ H) {
    const int tile = blockIdx.x * 4 + threadIdx.y;   // 4 waves per block
    const int tm   = tile / TILES_N;
    const int tn   = tile - tm * TILES_N;
    const int lane = threadIdx.x;                    // wave32
    const int half = lane >> 4;                      // 0: K{0,1}, 1: K{2,3}
    const int l16  = lane & 15;

    const float* __restrict__ xr = X + (size_t)(tm * 16 + l16) * DD; // A row
    const float* __restrict__ wc = W + tn * 16 + l16;                // B column

    v8f acc = {};
#pragma unroll
    for (int kk = 0; kk < DD; kk += 4) {
        const int k = kk + half * 2;                 // even -> 8B aligned
        v2f a = *(const v2f*)(xr + k);               // A[row][k], A[row][k+1]
        v2f b;
        b.x = wc[(size_t)k * DD];                    // B[k][n]
        b.y = wc[(size_t)(k + 1) * DD];              // B[k+1][n]
        acc = __builtin_amdgcn_wmma_f32_16x16x4_f32(
            /*neg_a=*/false, a, /*neg_b=*/false, b,
            /*c_mod=*/(short)0, acc, /*reuse_a=*/false, /*reuse_b=*/false);
    }

    // D layout: VGPR r -> row tm*16 + r + 8*half, col tn*16 + l16
    float* __restrict__ out = H + (size_t)(tm * 16 + 8 * half) * DD + tn * 16 + l16;
#pragma unroll
    for (int r = 0; r < 8; ++r) out[(size_t)r * DD] = acc[r];
}

// ---------------- self-loop + bias init: Agg = b + H * dis^2 ----------------
__global__ __launch_bounds__(256) void k_selfbias(const float* __restrict__ H,
                                                  const float* __restrict__ dis,
                                                  const float* __restrict__ b,
                                                  float* __restrict__ Agg, int total) {
    int t = blockIdx.x * blockDim.x + threadIdx.x;
    if (t >= total) return;
    int i = t / DD;
    int d = t - i * DD;
    float s = dis[i];
    Agg[t] = b[d] + H[t] * s * s;
}

// ---------------- edge scatter-add: Agg[dst] += H[src] * dis[src]*dis[dst] ----------------
// 32 lanes per edge, 3 floats per lane (96 = 32*3).
__global__ __launch_bounds__(256) void k_scatter(const float* __restrict__ H,
                                                 const int* __restrict__ src,
                                                 const int* __restrict__ dst,
                                                 const float* __restrict__ dis,
                                                 float* Agg, int e) {
    int t = blockIdx.x * blockDim.x + threadIdx.x;
    int eid = t >> 5;
    if (eid >= e) return;
    int lane = t & 31;
    int s = src[eid];
    int d = dst[eid];
    float nrm = dis[s] * dis[d];
    const float* __restrict__ hs = H + (size_t)s * DD + lane * 3;
    float* ad = Agg + (size_t)d * DD + lane * 3;
    atomicAdd(&ad[0], hs[0] * nrm);
    atomicAdd(&ad[1], hs[1] * nrm);
    atomicAdd(&ad[2], hs[2] * nrm);
}

// ---------------- ReLU ----------------
__global__ __launch_bounds__(256) void k_relu(const float* in, float* out, int total) {
    int t = blockIdx.x * blockDim.x + threadIdx.x;
    if (t < total) out[t] = fmaxf(in[t], 0.0f);
}

extern "C" void kernel_launch(void* const* d_in, const int* in_sizes, int n_in,
                              void* d_out, int out_size, void* d_ws, size_t ws_size,
                              hipStream_t stream) {
    const float* x   = (const float*)d_in[0];
    const int*   ei  = (const int*)d_in[1];   // [2, E]: row 0 = src, row 1 = dst
    const float* W1  = (const float*)d_in[2];
    const float* b1  = (const float*)d_in[3];
    const float* W2  = (const float*)d_in[4];
    const float* b2  = (const float*)d_in[5];
    float* out = (float*)d_out;

    const int* src = ei;
    const int* dst = ei + EE;

    // workspace: dis[N] | hbuf[N*96] | abuf[N*96]
    char* ws = (char*)d_ws;
    float* dis = (float*)ws;
    size_t off = ((size_t)NN * sizeof(float) + 255) & ~(size_t)255;
    float* hbuf = (float*)(ws + off);
    float* abuf = (float*)(ws + off + (size_t)NN * DD * sizeof(float));

    const int total = NN * DD;                 // 9,600,000
    const int scat_threads = EE * 32;          // 25,600,000
    dim3 gemm_block(32, 4);
    const int gemm_grid = TILES_TOTAL / 4;     // 9375

    // normalization: deg -> dis
    k_init_deg  <<<(NN + 255) / 256, 256, 0, stream>>>(dis, NN);
    k_deg_accum <<<(EE + 255) / 256, 256, 0, stream>>>(dst, dis, EE);
    k_rsqrt_inplace<<<(NN + 255) / 256, 256, 0, stream>>>(dis, NN);

    // layer 1: h = x@W1 ; agg = b1 + self + scatter ; a1 = relu(agg) -> hbuf
    k_gemm96_wmma<<<gemm_grid, gemm_block, 0, stream>>>(x, W1, hbuf);
    k_selfbias   <<<(total + 255) / 256, 256, 0, stream>>>(hbuf, dis, b1, abuf, total);
    k_scatter    <<<(scat_threads + 255) / 256, 256, 0, stream>>>(hbuf, src, dst, dis, abuf, EE);
    k_relu       <<<(total + 255) / 256, 256, 0, stream>>>(abuf, hbuf, total);

    // layer 2: h2 = a1@W2 -> abuf ; out = relu(b2 + self + scatter)
    k_gemm96_wmma<<<gemm_grid, gemm_block, 0, stream>>>(hbuf, W2, abuf);
    k_selfbias   <<<(total + 255) / 256, 256, 0, stream>>>(abuf, dis, b2, out, total);
    k_scatter    <<<(scat_threads + 255) / 256, 256, 0, stream>>>(abuf, src, dst, dis, out, EE);
    k_relu       <<<(total + 255) / 256, 256, 0, stream>>>(out, out, total);
}